// GatedNode2Edge_11605001634322
// MI455X (gfx1250) — compile-verified
//
#include <hip/hip_runtime.h>
#include <hip/hip_bf16.h>

// Shapes from the reference
#define B_DIM 2
#define C_DIM 64
#define N_DIM 1024
#define E_DIM 32

typedef float v2f __attribute__((ext_vector_type(2)));
typedef float v4f __attribute__((ext_vector_type(4)));
typedef float v8f __attribute__((ext_vector_type(8)));

// ---------------------------------------------------------------------------
// Kernel 1: t = th12 @ emb[b] for both parameter sets, via V_WMMA_F32_16X16X4_F32.
// One wave (32 threads) computes one 16x16 tile of t (M=E-tile, N=N-tile),
// accumulating over C=64 in 16 steps of K=4. The emb (B-matrix) tile is shared
// by both WMMA chains.
//
// VGPR layouts (ISA 7.12.2, f32 16x16x4):
//   A (16x4): lane 0-15 -> M=lane, v.x=K0 v.y=K1 ; lane 16-31 -> M=lane-16, v.x=K2 v.y=K3
//   B (4x16): lane 0-15 -> N=lane, v.x=K0 v.y=K1 ; lane 16-31 -> N=lane-16, v.x=K2 v.y=K3
//   C/D (16x16): VGPR g, lane l -> M = g + 8*(l>=16), N = l&15
// ---------------------------------------------------------------------------
__global__ __launch_bounds__(32) void n2e_wmma_gemm_kernel(
    const float* __restrict__ emb,     // (B, C, N)
    const float* __restrict__ th12_1,  // (E, C)
    const float* __restrict__ th12_2,  // (E, C)
    float* __restrict__ t1,            // (B, E, N)  in d_ws
    float* __restrict__ t2)            // (B, E, N)  in d_ws
{
    const int wid   = blockIdx.x;        // 0 .. B*(E/16)*(N/16)-1 = 255
    const int ntile = wid & (N_DIM / 16 - 1);          // 0..63
    const int etile = (wid >> 6) & (E_DIM / 16 - 1);   // 0..1
    const int b     = wid >> 7;                        // 0..1

    const int lane = threadIdx.x;        // 0..31
    const int half = lane >> 4;          // 0: K pair {0,1}; 1: K pair {2,3}
    const int l    = lane & 15;          // M (for A) / N (for B) / N (for C,D)
    const int e0   = etile * 16;
    const int n0   = ntile * 16;

    v8f acc1 = {};
    v8f acc2 = {};

    #pragma unroll
    for (int k = 0; k < C_DIM; k += 4) {
        const int klo = k + half * 2;

        v2f a1, a2, bm;
        a1.x = th12_1[(e0 + l) * C_DIM + klo];
        a1.y = th12_1[(e0 + l) * C_DIM + klo + 1];
        a2.x = th12_2[(e0 + l) * C_DIM + klo];
        a2.y = th12_2[(e0 + l) * C_DIM + klo + 1];
        bm.x = emb[((size_t)b * C_DIM + klo)     * N_DIM + n0 + l];
        bm.y = emb[((size_t)b * C_DIM + klo + 1) * N_DIM + n0 + l];

        // 8 args: (neg_a, A, neg_b, B, c_mod, C, reuse_a, reuse_b)
        acc1 = __builtin_amdgcn_wmma_f32_16x16x4_f32(
                   false, a1, false, bm, (short)0, acc1, false, false);
        acc2 = __builtin_amdgcn_wmma_f32_16x16x4_f32(
                   false, a2, false, bm, (short)0, acc2, false, false);
    }

    #pragma unroll
    for (int g = 0; g < 8; ++g) {
        const int m = g + 8 * half;
        t1[((size_t)b * E_DIM + e0 + m) * N_DIM + n0 + l] = acc1[g];
        t2[((size_t)b * E_DIM + e0 + m) * N_DIM + n0 + l] = acc2[g];
    }
}

// ---------------------------------------------------------------------------
// Kernel 2: streaming broadcast stage. One 256-thread block per (b,e,i) row.
// t rows (4 KB each) are L2/L1 resident and reused 1024x; the 256 MB output
// is written with 128-bit non-temporal stores (write-once data > L2 capacity).
// out[b,e,i,j] = relu(2*max(t1i,t1j) + th5_1[e]*(i==j))
//               * sigmoid(2*max(t2i,t2j) + th5_2[e]*(i==j))
// ---------------------------------------------------------------------------
__global__ __launch_bounds__(256) void n2e_fuse_kernel(
    const float* __restrict__ t1,     // (B, E, N)
    const float* __restrict__ t2,     // (B, E, N)
    const float* __restrict__ th5_1,  // (E,)
    const float* __restrict__ th5_2,  // (E,)
    float* __restrict__ out)          // (B, E, N, N)
{
    const int bei = blockIdx.x;              // 0 .. B*E*N-1
    const int i   = bei & (N_DIM - 1);
    const int be  = bei >> 10;               // b*E + e
    const int e   = be & (E_DIM - 1);

    const size_t rowbase = (size_t)be * N_DIM;
    const float  ti1 = t1[rowbase + i];
    const float  ti2 = t2[rowbase + i];
    const float  d1  = th5_1[e];
    const float  d2  = th5_2[e];

    const int j0 = threadIdx.x * 4;
    const v4f vj1 = *(const v4f*)(t1 + rowbase + j0);
    const v4f vj2 = *(const v4f*)(t2 + rowbase + j0);

    v4f r;
    #pragma unroll
    for (int k = 0; k < 4; ++k) {
        const int   j    = j0 + k;
        const float m1   = fmaxf(ti1, vj1[k]);
        const float m2   = fmaxf(ti2, vj2[k]);
        const float diag = (j == i) ? 1.0f : 0.0f;
        const float a    = fmaxf(m1 + m1 + d1 * diag, 0.0f);
        const float x    = m2 + m2 + d2 * diag;
        const float g    = 1.0f / (1.0f + __expf(-x));
        r[k] = a * g;
    }

    __builtin_nontemporal_store(r, (v4f*)(out + (size_t)bei * N_DIM + j0));
}

extern "C" void kernel_launch(void* const* d_in, const int* in_sizes, int n_in,
                              void* d_out, int out_size, void* d_ws, size_t ws_size,
                              hipStream_t stream) {
    // setup_inputs() order: emb, th12_1, th34_1, th5_1, th12_2, th34_2, th5_2
    const float* emb    = (const float*)d_in[0];
    const float* th12_1 = (const float*)d_in[1];
    const float* th5_1  = (const float*)d_in[3];
    const float* th12_2 = (const float*)d_in[4];
    const float* th5_2  = (const float*)d_in[6];
    float* out = (float*)d_out;

    float* t1 = (float*)d_ws;                         // B*E*N = 65536 floats
    float* t2 = t1 + (size_t)B_DIM * E_DIM * N_DIM;   // another 65536 floats

    // Kernel 1: 256 tiles, one wave each.
    n2e_wmma_gemm_kernel<<<B_DIM * (E_DIM / 16) * (N_DIM / 16), 32, 0, stream>>>(
        emb, th12_1, th12_2, t1, t2);

    // Kernel 2: one block per (b, e, i) row; 256 threads x float4 = 1024 cols.
    n2e_fuse_kernel<<<B_DIM * E_DIM * N_DIM, 256, 0, stream>>>(
        t1, t2, th5_1, th5_2, out);
}